// GraphAttentionLayer_3212635538069
// MI455X (gfx1250) — compile-verified
//
#include <hip/hip_runtime.h>
#include <hip/hip_bf16.h>

typedef __attribute__((ext_vector_type(2))) float v2f;
typedef __attribute__((ext_vector_type(8))) float v8f;

#define NODES   50000
#define EDGES   800000
#define FEAT    128
#define NEGSLOPE 0.2f

// ---------------------------------------------------------------------------
// Kernel 1: h = x @ W^T  via V_WMMA_F32_16X16X4_F32  (fp32, full precision)
// One wave computes a 16x16 tile of h. blockDim=256 (8 waves): wave w -> col
// tile w (8 col tiles cover 128 cols), blockIdx.x -> row tile (3125 tiles).
//
// f32 A-matrix 16x4 layout: lanes 0-15 hold (M=lane, K=k,k+1) in v[0],v[1];
// lanes 16-31 hold (M=lane-16, K=k+2,k+3). B 4x16 mirrors it with N=lane%16.
// C/D: VGPR g holds row M=g (lanes 0-15) / M=g+8 (lanes 16-31), col N=lane%16.
// ---------------------------------------------------------------------------
__global__ void gat_gemm_wmma(const float* __restrict__ x,
                              const float* __restrict__ W,
                              float* __restrict__ h) {
    const int lane    = threadIdx.x & 31;
    const int wave    = threadIdx.x >> 5;   // col tile 0..7
    const int i0      = blockIdx.x * 16;    // row tile base
    const int j0      = wave * 16;          // col tile base
    const int halfSel = lane >> 4;          // 0: K+0,K+1   1: K+2,K+3
    const int l16     = lane & 15;

    const float* xr = x + (size_t)(i0 + l16) * FEAT + 2 * halfSel;
    const float* wr = W + (size_t)(j0 + l16) * FEAT + 2 * halfSel;

    v8f acc = {};
#pragma unroll
    for (int k = 0; k < FEAT; k += 4) {
        v2f a = *(const v2f*)(xr + k);   // x[i0+l16][k + 2*halfSel ..]
        v2f b = *(const v2f*)(wr + k);   // W[j0+l16][k + 2*halfSel ..]
        acc = __builtin_amdgcn_wmma_f32_16x16x4_f32(
            /*neg_a=*/false, a, /*neg_b=*/false, b,
            /*c_mod=*/(short)0, acc, /*reuse_a=*/false, /*reuse_b=*/false);
    }

    float* hp = h + (size_t)(i0 + 8 * halfSel) * FEAT + j0 + l16;
#pragma unroll
    for (int g = 0; g < 8; ++g)
        hp[(size_t)g * FEAT] = acc[g];
}

// ---------------------------------------------------------------------------
// Kernel 2: per-node attention scores s_src = h . a[0:128], s_dst = h . a[128:256]
// One wave32 per node; lane handles 4 features (float4), shfl reduction.
// ---------------------------------------------------------------------------
__global__ void gat_node_scores(const float* __restrict__ h,
                                const float* __restrict__ a,
                                float* __restrict__ s_src,
                                float* __restrict__ s_dst) {
    const int node = blockIdx.x * 8 + (threadIdx.x >> 5);
    const int lane = threadIdx.x & 31;
    if (node >= NODES) return;

    const float4 hv = *(const float4*)(h + (size_t)node * FEAT + lane * 4);
    const float4 a0 = *(const float4*)(a + lane * 4);
    const float4 a1 = *(const float4*)(a + FEAT + lane * 4);

    float ss = hv.x * a0.x + hv.y * a0.y + hv.z * a0.z + hv.w * a0.w;
    float sd = hv.x * a1.x + hv.y * a1.y + hv.z * a1.z + hv.w * a1.w;
#pragma unroll
    for (int off = 16; off > 0; off >>= 1) {
        ss += __shfl_down(ss, off, 32);
        sd += __shfl_down(sd, off, 32);
    }
    if (lane == 0) { s_src[node] = ss; s_dst[node] = sd; }
}

// ---------------------------------------------------------------------------
// Kernel 3: zero-init out, row_max (=0 matches torch clamp-at-0), row_sum.
// ---------------------------------------------------------------------------
__global__ void gat_init(float* __restrict__ out,
                         float* __restrict__ row_max,
                         float* __restrict__ row_sum) {
    const int idx = blockIdx.x * 256 + threadIdx.x;
    if (idx < NODES * FEAT) out[idx] = 0.0f;
    if (idx < NODES) { row_max[idx] = 0.0f; row_sum[idx] = 0.0f; }
}

// ---------------------------------------------------------------------------
// Kernel 4: edge logits e = leaky_relu(s_src[src]+s_dst[dst]); segment max.
// row_max starts at 0; only positive values can raise it, and for positive
// IEEE floats unsigned-int order == float order -> atomicMax on uint bits.
// ---------------------------------------------------------------------------
__global__ void gat_edge_logits(const int* __restrict__ src,
                                const int* __restrict__ dst,
                                const float* __restrict__ s_src,
                                const float* __restrict__ s_dst,
                                float* __restrict__ ebuf,
                                unsigned int* __restrict__ row_max_u) {
    const int e = blockIdx.x * 256 + threadIdx.x;
    if (e >= EDGES) return;
    const int s = src[e], d = dst[e];
    float v = s_src[s] + s_dst[d];
    v = (v > 0.0f) ? v : NEGSLOPE * v;
    ebuf[e] = v;
    if (v > 0.0f) atomicMax(row_max_u + s, __float_as_uint(v));
}

// ---------------------------------------------------------------------------
// Kernel 5: v = exp(e - row_max[src]); segment sum via f32 atomics.
// ---------------------------------------------------------------------------
__global__ void gat_edge_exp(const int* __restrict__ src,
                             const float* __restrict__ row_max,
                             float* __restrict__ ebuf,
                             float* __restrict__ row_sum) {
    const int e = blockIdx.x * 256 + threadIdx.x;
    if (e >= EDGES) return;
    const int s = src[e];
    const float v = __expf(ebuf[e] - row_max[s]);
    ebuf[e] = v;
    atomicAdd(row_sum + s, v);
}

// ---------------------------------------------------------------------------
// Kernel 6: out[src] += (v / row_sum[src]) * h[dst]. One wave32 per edge;
// lane gathers float4 of h[dst] (coalesced 512B/edge, L2-resident) and does
// 4 global f32 atomic adds into out[src].
// ---------------------------------------------------------------------------
__global__ void gat_edge_scatter(const int* __restrict__ src,
                                 const int* __restrict__ dst,
                                 const float* __restrict__ ebuf,
                                 const float* __restrict__ row_sum,
                                 const float* __restrict__ h,
                                 float* __restrict__ out) {
    const int e    = blockIdx.x * 8 + (threadIdx.x >> 5);
    const int lane = threadIdx.x & 31;
    if (e >= EDGES) return;
    const int s = src[e], d = dst[e];
    const float rs   = row_sum[s];
    const float coef = ebuf[e] / (rs == 0.0f ? 1.0f : rs);

    const float4 hv = *(const float4*)(h + (size_t)d * FEAT + lane * 4);
    float* op = out + (size_t)s * FEAT + lane * 4;
    atomicAdd(op + 0, coef * hv.x);
    atomicAdd(op + 1, coef * hv.y);
    atomicAdd(op + 2, coef * hv.z);
    atomicAdd(op + 3, coef * hv.w);
}

// ---------------------------------------------------------------------------
extern "C" void kernel_launch(void* const* d_in, const int* in_sizes, int n_in,
                              void* d_out, int out_size, void* d_ws, size_t ws_size,
                              hipStream_t stream) {
    const float* x  = (const float*)d_in[0];          // [N,128]
    const int*   ei = (const int*)d_in[1];            // [2,E] flat
    const float* W  = (const float*)d_in[2];          // [128,128]
    const float* a  = (const float*)d_in[3];          // [256]
    float* out = (float*)d_out;                       // [N,128]

    const int* src = ei;           // row (softmax segments)
    const int* dst = ei + EDGES;

    // workspace layout (floats)
    float* ws      = (float*)d_ws;
    float* h       = ws;                               // N*128
    float* s_src   = h + (size_t)NODES * FEAT;         // N
    float* s_dst   = s_src + NODES;                    // N
    float* ebuf    = s_dst + NODES;                    // E
    float* row_max = ebuf + EDGES;                     // N
    float* row_sum = row_max + NODES;                  // N

    // 1) h = x @ W^T  (WMMA f32)
    gat_gemm_wmma<<<NODES / 16, 256, 0, stream>>>(x, W, h);

    // 2) node scores
    gat_node_scores<<<NODES / 8, 256, 0, stream>>>(h, a, s_src, s_dst);

    // 3) zero-init out / row_max / row_sum
    gat_init<<<(NODES * FEAT + 255) / 256, 256, 0, stream>>>(out, row_max, row_sum);

    // 4) edge logits + segment max
    gat_edge_logits<<<(EDGES + 255) / 256, 256, 0, stream>>>(
        src, dst, s_src, s_dst, ebuf, (unsigned int*)row_max);

    // 5) exp + segment sum
    gat_edge_exp<<<(EDGES + 255) / 256, 256, 0, stream>>>(src, row_max, ebuf, row_sum);

    // 6) normalize + weighted scatter-add of neighbor features
    gat_edge_scatter<<<EDGES / 8, 256, 0, stream>>>(src, dst, ebuf, row_sum, h, out);
}